// SGConvNet_51754355916838
// MI455X (gfx1250) — compile-verified
//
#include <hip/hip_runtime.h>
#include <hip/hip_bf16.h>
#include <math.h>

// Problem constants (match reference)
#define N_NODES 50000
#define N_EDGES 800000
#define F_IN    128
#define HID     160
#define OUT_C   128
#define NGR     64
#define OUT_DIM 10
#define K_HOPS  5

#define SCAN_B   256
#define N_BLKS   ((N_NODES + SCAN_B - 1) / SCAN_B)   // 196 <= 256
#define TILE_PAD 20   // LDS row stride in floats: 80B = 16B-aligned, conflict-free

typedef __attribute__((ext_vector_type(2))) float v2f;
typedef __attribute__((ext_vector_type(8))) float v8f;

// ---------------- utility fills ----------------
__global__ void fill_f32(float* __restrict__ p, float v, int n) {
    int i = blockIdx.x * blockDim.x + threadIdx.x;
    if (i < n) p[i] = v;
}
__global__ void fill_i32(int* __restrict__ p, int v, int n) {
    int i = blockIdx.x * blockDim.x + threadIdx.x;
    if (i < n) p[i] = v;
}

// ---------------- degree / dinv ----------------
__global__ void count_deg(const int* __restrict__ dst, int* __restrict__ deg, int e) {
    int i = blockIdx.x * blockDim.x + threadIdx.x;
    if (i < e) atomicAdd(&deg[dst[i]], 1);
}
// deg here excludes the self-loop; reference deg includes it (+1)
__global__ void make_dinv(const int* __restrict__ deg, float* __restrict__ dinv, int n) {
    int i = blockIdx.x * blockDim.x + threadIdx.x;
    if (i < n) dinv[i] = 1.0f / sqrtf((float)(deg[i] + 1));
}

// ---------------- CSR build: exclusive scan of degrees ----------------
__global__ void scan_block(const int* __restrict__ deg, int* __restrict__ incl,
                           int* __restrict__ bsum, int n) {
    __shared__ int s[SCAN_B];
    int i = blockIdx.x * SCAN_B + threadIdx.x;
    s[threadIdx.x] = (i < n) ? deg[i] : 0;
    __syncthreads();
    for (int off = 1; off < SCAN_B; off <<= 1) {
        int t = (threadIdx.x >= off) ? s[threadIdx.x - off] : 0;
        __syncthreads();
        s[threadIdx.x] += t;
        __syncthreads();
    }
    if (i < n) incl[i] = s[threadIdx.x];
    if (threadIdx.x == SCAN_B - 1) bsum[blockIdx.x] = s[SCAN_B - 1];
}
__global__ void scan_sums(int* __restrict__ bsum, int nb) {   // serial exclusive scan (nb<=256)
    if (threadIdx.x == 0) {
        int run = 0;
        for (int b = 0; b < nb; ++b) { int t = bsum[b]; bsum[b] = run; run += t; }
    }
}
__global__ void scan_finish(const int* __restrict__ deg, const int* __restrict__ incl,
                            const int* __restrict__ bsum, int* __restrict__ rowstart,
                            int n, int total) {
    int i = blockIdx.x * blockDim.x + threadIdx.x;
    if (i < n) rowstart[i] = incl[i] - deg[i] + bsum[i >> 8];
    if (i == n) rowstart[n] = total;
}
__global__ void csr_scatter(const int* __restrict__ src, const int* __restrict__ dst,
                            int* __restrict__ pos, int* __restrict__ csr_src, int e) {
    int i = blockIdx.x * blockDim.x + threadIdx.x;
    if (i < e) {
        int p = atomicAdd(&pos[dst[i]], 1);
        csr_src[p] = src[i];
    }
}

// ---------------- propagation hop (atomic-free CSR gather) ----------------
// out[i,f] = dinv[i]^2 * in[i,f] + sum_{s in N(i)} dinv[s]*dinv[i] * in[s,f]
__global__ void prop_csr(const float* __restrict__ in, const float* __restrict__ dinv,
                         const int* __restrict__ rowstart, const int* __restrict__ csr_src,
                         float* __restrict__ out, int F) {
    int i = blockIdx.x;        // node
    int f = threadIdx.x;       // feature
    float d = dinv[i];
    float acc = in[(size_t)i * F + f] * d * d;          // self-loop term
    int e0 = rowstart[i];
    int e1 = rowstart[i + 1];
    for (int e = e0; e < e1; ++e) {
        int s = csr_src[e];                              // block-uniform -> scalar load
        acc += in[(size_t)s * F + f] * (dinv[s] * d);    // coalesced row gather
    }
    out[(size_t)i * F + f] = acc;
}

// ---------------- WMMA fp32 GEMM: C = A[M x K] @ W[K x N] + bias ----------------
// One wave32 per 16x16 tile. A tile staged to LDS via CDNA5 async-to-LDS DMA.
// M=50000, K in {128,160}, N in {160,128}: all multiples of tile sizes,
// so EXEC stays all-ones around every WMMA (ISA requirement).
__global__ __launch_bounds__(32)
void gemm_wmma_f32(const float* __restrict__ A, const float* __restrict__ W,
                   const float* __restrict__ bias, float* __restrict__ C,
                   int Kdim, int Ndim) {
    __shared__ __align__(16) float tile[16 * TILE_PAD];   // 16 rows x 16 cols (+4 pad)

    const int lane = threadIdx.x;      // 0..31
    const int half = lane >> 4;        // 0: lanes 0-15, 1: lanes 16-31
    const int l15  = lane & 15;
    const int m0 = blockIdx.x * 16;
    const int n0 = blockIdx.y * 16;
    const int n  = n0 + l15;

    v8f acc = {};

    // Staging map: lane (r=l15, h=half) copies row r, cols [4h,4h+4) and [8+4h,8+4h+4).
    // INST_OFFSET is added to BOTH the LDS and global addresses (ISA 08 pseudocode),
    // so the second b128 is the same operands with offset:32.
    const float* arow = A + (size_t)(m0 + l15) * Kdim + half * 4;
    unsigned lds_addr = (unsigned)(size_t)&tile[l15 * TILE_PAD + half * 4];

    for (int k0 = 0; k0 < Kdim; k0 += 16) {
        unsigned long long ga = (unsigned long long)(size_t)(arow + k0);
        // previous iteration's ds reads must retire before DMA overwrites the tile
        asm volatile("s_wait_dscnt 0x0" ::: "memory");
        asm volatile(
            "global_load_async_to_lds_b128 %0, %1, off\n\t"
            "global_load_async_to_lds_b128 %0, %1, off offset:32\n\t"
            "s_wait_asynccnt 0x0"
            :: "v"(lds_addr), "v"(ga)
            : "memory");

#pragma unroll
        for (int kk = 0; kk < 16; kk += 4) {
            v2f a, b;
            // A 16x4 frag: row = lane&15; VGPR0 = K{0|2}, VGPR1 = K{1|3}
            a.x = tile[l15 * TILE_PAD + kk + 2 * half + 0];
            a.y = tile[l15 * TILE_PAD + kk + 2 * half + 1];
            // B 4x16 frag: col = lane&15; low lanes K=0..1, high lanes K=2..3
            b.x = W[(size_t)(k0 + kk + 2 * half + 0) * Ndim + n];
            b.y = W[(size_t)(k0 + kk + 2 * half + 1) * Ndim + n];
            acc = __builtin_amdgcn_wmma_f32_16x16x4_f32(
                /*neg_a=*/false, a, /*neg_b=*/false, b,
                /*c_mod=*/(short)0, acc, /*reuse_a=*/false, /*reuse_b=*/false);
        }
    }

    // D 16x16 f32: VGPR v -> row m0 + v + 8*half, col n0 + (lane&15)
    float bc = bias[n];
#pragma unroll
    for (int v = 0; v < 8; ++v) {
        C[(size_t)(m0 + v + 8 * half) * Ndim + n] = acc[v] + bc;
    }
}

// ---------------- global mean pool ----------------
__global__ void pool_accum(const float* __restrict__ h, const int* __restrict__ batch,
                           float* __restrict__ pooled, float* __restrict__ counts) {
    int i = blockIdx.x;       // node
    int f = threadIdx.x;      // 0..OUT_C-1
    int g = batch[i];
    atomicAdd(&pooled[(size_t)g * OUT_C + f], h[(size_t)i * OUT_C + f]);
    if (f == 0) atomicAdd(&counts[g], 1.0f);
}

// ---------------- FC + log_softmax (64 graphs x 10 classes) ----------------
__global__ void fc_logsoftmax(const float* __restrict__ pooled, const float* __restrict__ counts,
                              const float* __restrict__ fc_w, const float* __restrict__ fc_b,
                              float* __restrict__ out) {
    int g = threadIdx.x;
    if (g >= NGR) return;
    float c = counts[g];
    c = (c > 1.0f) ? c : 1.0f;
    float logits[OUT_DIM];
#pragma unroll
    for (int d = 0; d < OUT_DIM; ++d) {
        float s = fc_b[d];
        for (int k = 0; k < OUT_C; ++k)
            s += (pooled[(size_t)g * OUT_C + k] / c) * fc_w[k * OUT_DIM + d];
        logits[d] = s;
    }
    float mx = logits[0];
#pragma unroll
    for (int d = 1; d < OUT_DIM; ++d) mx = fmaxf(mx, logits[d]);
    float se = 0.0f;
#pragma unroll
    for (int d = 0; d < OUT_DIM; ++d) se += expf(logits[d] - mx);
    float lse = mx + logf(se);
#pragma unroll
    for (int d = 0; d < OUT_DIM; ++d) out[g * OUT_DIM + d] = logits[d] - lse;
}

extern "C" void kernel_launch(void* const* d_in, const int* in_sizes, int n_in,
                              void* d_out, int out_size, void* d_ws, size_t ws_size,
                              hipStream_t stream) {
    const float* x    = (const float*)d_in[0];
    const int*   eidx = (const int*)d_in[1];   // [2, E]: row0 = src, row1 = dst
    const int*   batch= (const int*)d_in[2];
    const float* W1   = (const float*)d_in[3];
    const float* b1   = (const float*)d_in[4];
    const float* W2   = (const float*)d_in[5];
    const float* b2   = (const float*)d_in[6];
    const float* W3   = (const float*)d_in[7];
    const float* b3   = (const float*)d_in[8];
    const float* fcw  = (const float*)d_in[9];
    const float* fcb  = (const float*)d_in[10];
    float* out = (float*)d_out;

    const int* src = eidx;
    const int* dst = eidx + N_EDGES;

    // carve workspace
    char* ws = (char*)d_ws;
    size_t off = 0;
    auto carve = [&](size_t bytes) -> void* {
        void* p = ws + off;
        off += (bytes + 255) & ~(size_t)255;
        return p;
    };
    float* dinv    = (float*)carve((size_t)N_NODES * sizeof(float));
    float* bufA    = (float*)carve((size_t)N_NODES * HID * sizeof(float));
    float* bufB    = (float*)carve((size_t)N_NODES * HID * sizeof(float));
    float* pooled  = (float*)carve((size_t)NGR * OUT_C * sizeof(float));
    float* counts  = (float*)carve((size_t)NGR * sizeof(float));
    int*   degi    = (int*)carve((size_t)N_NODES * sizeof(int));
    int*   incl    = (int*)carve((size_t)N_NODES * sizeof(int));
    int*   bsum    = (int*)carve((size_t)SCAN_B * sizeof(int));
    int*   rowstart= (int*)carve((size_t)(N_NODES + 1) * sizeof(int));
    int*   pos     = (int*)carve((size_t)N_NODES * sizeof(int));
    int*   csr_src = (int*)carve((size_t)N_EDGES * sizeof(int));

    // ---- degree count + dinv (deg includes self-loop) ----
    fill_i32<<<(N_NODES + 255) / 256, 256, 0, stream>>>(degi, 0, N_NODES);
    count_deg<<<(N_EDGES + 255) / 256, 256, 0, stream>>>(dst, degi, N_EDGES);
    make_dinv<<<(N_NODES + 255) / 256, 256, 0, stream>>>(degi, dinv, N_NODES);

    // ---- CSR build (once per launch; edges are static within a call) ----
    scan_block<<<N_BLKS, SCAN_B, 0, stream>>>(degi, incl, bsum, N_NODES);
    scan_sums<<<1, 32, 0, stream>>>(bsum, N_BLKS);
    scan_finish<<<(N_NODES + 256) / 256, 256, 0, stream>>>(degi, incl, bsum, rowstart,
                                                           N_NODES, N_EDGES);
    // pos = rowstart[0..N)
    hipMemcpyAsync(pos, rowstart, (size_t)N_NODES * sizeof(int),
                   hipMemcpyDeviceToDevice, stream);
    csr_scatter<<<(N_EDGES + 255) / 256, 256, 0, stream>>>(src, dst, pos, csr_src, N_EDGES);

    auto prop = [&](const float* in, float* o, int F) {
        prop_csr<<<N_NODES, F, 0, stream>>>(in, dinv, rowstart, csr_src, o, F);
    };
    auto gemm = [&](const float* A, const float* W, const float* b, float* C,
                    int Kd, int Nd) {
        gemm_wmma_f32<<<dim3(N_NODES / 16, Nd / 16), 32, 0, stream>>>(A, W, b, C, Kd, Nd);
    };

    // ---- SGConv layer 1 (F=128 -> 160) ----
    prop(x,    bufA, F_IN);
    prop(bufA, bufB, F_IN);
    prop(bufB, bufA, F_IN);
    prop(bufA, bufB, F_IN);
    prop(bufB, bufA, F_IN);
    gemm(bufA, W1, b1, bufB, F_IN, HID);      // bufB: N x 160

    // ---- SGConv layer 2 (F=160 -> 160) ----
    prop(bufB, bufA, HID);
    prop(bufA, bufB, HID);
    prop(bufB, bufA, HID);
    prop(bufA, bufB, HID);
    prop(bufB, bufA, HID);
    gemm(bufA, W2, b2, bufB, HID, HID);       // bufB: N x 160

    // ---- SGConv layer 3 (F=160 -> 128) ----
    prop(bufB, bufA, HID);
    prop(bufA, bufB, HID);
    prop(bufB, bufA, HID);
    prop(bufA, bufB, HID);
    prop(bufB, bufA, HID);
    gemm(bufA, W3, b3, bufB, HID, OUT_C);     // bufB: N x 128

    // ---- global mean pool + FC + log_softmax ----
    fill_f32<<<(NGR * OUT_C + 255) / 256, 256, 0, stream>>>(pooled, 0.0f, NGR * OUT_C);
    fill_f32<<<1, 64, 0, stream>>>(counts, 0.0f, NGR);
    pool_accum<<<N_NODES, OUT_C, 0, stream>>>(bufB, batch, pooled, counts);
    fc_logsoftmax<<<1, 64, 0, stream>>>(pooled, counts, fcw, fcb, out);
}